// PNNLayer_77378130805153
// MI455X (gfx1250) — compile-verified
//
#include <hip/hip_runtime.h>

// ---------------------------------------------------------------------------
// PNN layer, algebraically collapsed:
//   out[N,64] = X[N,96] @ Wc[96,64] + b,   X = [dists | embeds],
//   Wc = [ (embeds[anchors] @ W1) / 32 ; W2 ]
// Main GEMM runs on V_WMMA_F32_16X16X4_F32 (full f32 precision).
// Wc is held in LDS pair-interleaved so each B fragment is ONE ds_load_b64
// into an even-aligned VGPR pair (no re-pairing movs before the WMMAs).
// ---------------------------------------------------------------------------

typedef float v2f __attribute__((ext_vector_type(2)));
typedef float v4f __attribute__((ext_vector_type(4)));
typedef float v8f __attribute__((ext_vector_type(8)));

#define N_NODES   100000
#define N_ANCH    32
#define EMB       64
#define KTOT      96            // 32 (dists) + 64 (embeds)
#define KSTEPS    (KTOT / 4)    // 24 wmma K-steps
#define XSTRIDE   100           // LDS row stride (floats) for X tile, conflict-free
#define WPSTR     160           // LDS pair-row stride (floats): 160 % 64 == 32
                                //  -> lanes 16..31 hit the opposite 32 banks
#define NODES_PB  64            // nodes per block (4 waves x 16)
#define THREADS   128           // 4 waves (wave32)

// ---------------------------------------------------------------------------
// Kernel 1: build Wc[96][64] in workspace.
//   rows 0..31 :  (1/32) * embeds[anchor[r]] @ W_hidden[0:64]
//   rows 32..95:  W_hidden[64 + (r-32)]
// ---------------------------------------------------------------------------
__global__ void pnn_prep_wc(const float* __restrict__ embeds,
                            const float* __restrict__ W,      // [128][64]
                            const int*   __restrict__ anchors,
                            float* __restrict__ Wc)           // [96][64]
{
    int idx = blockIdx.x * blockDim.x + threadIdx.x;
    if (idx >= KTOT * EMB) return;
    int r = idx / EMB;
    int d = idx - r * EMB;
    if (r < N_ANCH) {
        const float* erow = embeds + (long)anchors[r] * EMB;
        float s = 0.0f;
        #pragma unroll
        for (int e = 0; e < EMB; ++e)
            s = fmaf(erow[e], W[e * EMB + d], s);
        Wc[idx] = s * (1.0f / (float)N_ANCH);
    } else {
        Wc[idx] = W[(EMB + (r - N_ANCH)) * EMB + d];
    }
}

// ---------------------------------------------------------------------------
// Kernel 2: out = X @ Wc + b via f32 WMMA.
// Each wave computes a 16-node x 64-col tile: 24 K-steps x 4 N-tiles.
// ---------------------------------------------------------------------------
__global__ __launch_bounds__(THREADS)
void pnn_fused_wmma(const float* __restrict__ embeds,
                    const float* __restrict__ dists,
                    const float* __restrict__ bias,
                    const float* __restrict__ Wc,
                    float* __restrict__ out)
{
    __shared__ float Xs[NODES_PB * XSTRIDE];        // 25600 B
    __shared__ float Wp[(KTOT / 2) * WPSTR];        // 30720 B, pair-interleaved

    const int tid = threadIdx.x;
    const int m0  = blockIdx.x * NODES_PB;

    // ---- stage Wc pair-interleaved: Wp[pr][c] = {Wc[2pr][c], Wc[2pr+1][c]} ----
    for (int i = tid; i < (KTOT / 2) * EMB; i += THREADS) {
        int pr = i >> 6;            // / EMB
        int c  = i & (EMB - 1);
        v2f v;
        v.x = Wc[(2 * pr + 0) * EMB + c];
        v.y = Wc[(2 * pr + 1) * EMB + c];
        *(v2f*)&Wp[pr * WPSTR + c * 2] = v;
    }
    // ---- stage X tile: dists -> cols [0,32), embeds -> cols [32,96) ----
    for (int i = tid; i < NODES_PB * (N_ANCH / 4); i += THREADS) {
        int node = i / (N_ANCH / 4);
        int c4   = i - node * (N_ANCH / 4);
        int g = m0 + node; if (g >= N_NODES) g = N_NODES - 1;   // clamp tail
        v4f v = ((const v4f*)dists)[g * (N_ANCH / 4) + c4];
        *(v4f*)&Xs[node * XSTRIDE + c4 * 4] = v;
    }
    for (int i = tid; i < NODES_PB * (EMB / 4); i += THREADS) {
        int node = i / (EMB / 4);
        int c4   = i - node * (EMB / 4);
        int g = m0 + node; if (g >= N_NODES) g = N_NODES - 1;   // clamp tail
        v4f v = ((const v4f*)embeds)[g * (EMB / 4) + c4];
        *(v4f*)&Xs[node * XSTRIDE + N_ANCH + c4 * 4] = v;
    }
    __syncthreads();

    const int lane = tid & 31;
    const int wave = tid >> 5;
    const int l16  = lane & 15;
    const int half = lane >> 4;                 // 0: K=4k,4k+1   1: K=4k+2,4k+3
    const int rowA = wave * 16 + l16;           // local node row of A fragment
    const float* xrow = &Xs[rowA * XSTRIDE + 2 * half];
    // B fragment base: pair-row pr = 2k + half, column (n*16 + l16)
    const float* wbase = &Wp[half * WPSTR + l16 * 2];

    // C init = bias broadcast: lane covers column n*16 + l16 for all 8 M rows
    v8f acc[4];
    #pragma unroll
    for (int n = 0; n < 4; ++n) {
        float bv = bias[n * 16 + l16];
        v8f a8;
        #pragma unroll
        for (int j = 0; j < 8; ++j) a8[j] = bv;
        acc[n] = a8;
    }

    #pragma unroll
    for (int k = 0; k < KSTEPS; ++k) {
        // A fragment (16x4 f32): one ds_load_b64, conflict-free via XSTRIDE
        v2f a = *(const v2f*)&xrow[4 * k];
        #pragma unroll
        for (int n = 0; n < 4; ++n) {
            // B fragment: one ds_load_b64 straight into an aligned pair
            v2f bf = *(const v2f*)&wbase[2 * k * WPSTR + n * 32];
            acc[n] = __builtin_amdgcn_wmma_f32_16x16x4_f32(
                         false, a, false, bf, (short)0, acc[n], false, false);
        }
    }

    // ---- store D: vgpr i -> M = half*8 + i, col = n*16 + l16 ----
    #pragma unroll
    for (int n = 0; n < 4; ++n) {
        #pragma unroll
        for (int i = 0; i < 8; ++i) {
            int m = m0 + wave * 16 + half * 8 + i;
            if (m < N_NODES)
                out[(long)m * EMB + n * 16 + l16] = acc[n][i];
        }
    }
}

// ---------------------------------------------------------------------------
extern "C" void kernel_launch(void* const* d_in, const int* in_sizes, int n_in,
                              void* d_out, int out_size, void* d_ws, size_t ws_size,
                              hipStream_t stream) {
    const float* embeds = (const float*)d_in[0];   // [100000,64]
    const float* dists  = (const float*)d_in[1];   // [100000,32]
    const float* W      = (const float*)d_in[2];   // [128,64]
    const float* bias   = (const float*)d_in[3];   // [64]
    const int*   anch   = (const int*)d_in[4];     // [32]
    float* out = (float*)d_out;                    // [100000,64]
    float* Wc  = (float*)d_ws;                     // 96*64 f32 = 24.6 KB scratch

    (void)in_sizes; (void)n_in; (void)out_size; (void)ws_size;

    pnn_prep_wc<<<(KTOT * EMB + 255) / 256, 256, 0, stream>>>(embeds, W, anch, Wc);

    int nblocks = (N_NODES + NODES_PB - 1) / NODES_PB;
    pnn_fused_wmma<<<nblocks, THREADS, 0, stream>>>(embeds, dists, bias, Wc, out);
}